// Yolo_30717606101267
// MI455X (gfx1250) — compile-verified
//
#include <hip/hip_runtime.h>
#include <math.h>

// Problem constants (match reference)
#define S26    26
#define C80    80
#define NCH    255          // 3*(5+80)
#define CELLS  676          // 26*26
#define NB32   32
#define NIMG   256

typedef float v2f __attribute__((ext_vector_type(2)));
typedef float v8f __attribute__((ext_vector_type(8)));

__constant__ float ANCHOR_W[9] = {10.f,16.f,33.f,30.f,62.f,59.f,116.f,156.f,373.f};
__constant__ float ANCHOR_H[9] = {13.f,30.f,23.f,61.f,45.f,119.f,90.f,198.f,326.f};

// Exact-f32 wave32 sum using V_WMMA_F32_16X16X4_F32 with an all-ones B.
// A[m,k]: lane m supplies K=0 (a[0]) / K=1 (a[1]); lane 16+m supplies K=2/K=3.
// With a[1]=0 and B==1, D[m,n] = v[lane m] + v[lane 16+m] for every n.
// Summing D's 8 VGPRs gives rows 0..7 at lanes 0..15 and rows 8..15 at
// lanes 16..31; total = lane0 + lane16. All adds are f32 (RNE), exact.
// Must be called with EXEC all ones (uniform control flow across the block).
__device__ inline float wave_sum_wmma(float v) {
  v2f a; a[0] = v;   a[1] = 0.0f;
  v2f b; b[0] = 1.0f; b[1] = 1.0f;
  v8f c = {};
  c = __builtin_amdgcn_wmma_f32_16x16x4_f32(
      /*neg_a=*/false, a, /*neg_b=*/false, b,
      /*c_mod=*/(short)0, c, /*reuse_a=*/false, /*reuse_b=*/false);
  float s = c[0] + c[1] + c[2] + c[3] + c[4] + c[5] + c[6] + c[7];
  return __shfl(s, 0, 32) + __shfl(s, 16, 32);
}

// One block per image, 256 threads (8 wave32).
__global__ __launch_bounds__(256) void yolo_image_loss(
    const float* __restrict__ X,     // (256, 255, 26, 26)
    const float* __restrict__ BOX,   // (256, 32, 5) : cls, cx, cy, w, h
    const int*   __restrict__ IDX,   // (256, 32)    : anchor index in [3,5]
    float* __restrict__ per_img)     // (256,) per-image loss
{
  const int img = blockIdx.x;
  const int tid = threadIdx.x;
  const float* __restrict__ x = X + (size_t)img * (NCH * CELLS);

  __shared__ int   s_key[NB32];   // base*676 + cell  (obj-channel flat offset)
  __shared__ int   s_cls[NB32];
  __shared__ int   s_uni[NB32];
  __shared__ float s_ws[8];

  float boxterm = 0.0f;   // per-box coord/obj terms (threads 0..31)
  float corr    = 0.0f;   // unique obj^2 to remove from noobj sum
  float obj     = 0.0f;

  // ---- Phase 1: per-box gather + math (one box per lane of wave 0) ----
  if (tid < NB32) {
    const float* bb = BOX + ((size_t)img * NB32 + tid) * 5;
    const float cx = bb[1], cy = bb[2], bw = bb[3], bh = bb[4];
    const int nidx = IDX[(size_t)img * NB32 + tid];       // 3..5
    const int base = (nidx - 3) * 85;                     // 0 / 85 / 170
    const float div = 16.0f;                              // 416/26
    const int ix = (int)(cx / div);
    const int iy = (int)(cy / div);
    const float ax = (cx - (float)ix * div) / div;
    const float ay = (cy - (float)iy * div) / div;
    const int cell = ix * S26 + iy;
    const int key  = base * CELLS + cell;
    s_key[tid] = key;
    s_cls[tid] = (int)bb[0];

    obj            = x[key];                 // obj_pred
    const float rax = x[key + 1 * CELLS];    // raw (no sigmoid), per reference
    const float ray = x[key + 2 * CELLS];
    const float tw  = x[key + 3 * CELLS];
    const float th  = x[key + 4 * CELLS];
    const float sw  = 1.0f / (1.0f + expf(-tw));
    const float sh  = 1.0f / (1.0f + expf(-th));
    const float rw  = ANCHOR_W[nidx] * expf(4.0f * sw - 2.0f);
    const float rh  = ANCHOR_H[nidx] * expf(4.0f * sh - 2.0f);

    // IoU with the reference's +1 convention
    const float b1x1 = rax * div - rw * 0.5f, b1y1 = ray * div - rh * 0.5f;
    const float b1x2 = rax * div + rw * 0.5f, b1y2 = ray * div + rh * 0.5f;
    const float b2x1 = ax * div - bw * 0.5f,  b2y1 = ay * div - bh * 0.5f;
    const float b2x2 = ax * div + bw * 0.5f,  b2y2 = ay * div + bh * 0.5f;
    const float A  = (b1x2 - b1x1 + 1.0f) * (b1y2 - b1y1 + 1.0f);
    const float Bt = (b2x2 - b2x1 + 1.0f) * (b2y2 - b2y1 + 1.0f);
    const float CM = (fminf(b1x2, b2x2) - fmaxf(b1x1, b2x1) + 1.0f) *
                     (fminf(b1y2, b2y2) - fmaxf(b1y1, b2y1) + 1.0f);
    float r = CM / (A + Bt - CM);
    const float iou = (r < 0.0f) ? 0.0f : r;

    const float d0 = obj - iou;
    const float d1 = rax - ax;
    const float d2 = ray - ay;
    const float d3 = rw / 416.0f - bw / 416.0f;
    const float d4 = rh / 416.0f - bh / 416.0f;
    boxterm = d0*d0 + d1*d1 + d2*d2 + d3*d3 + d4*d4;   // scaled by 5/32 later
  }
  __syncthreads();

  // ---- Phase 2: dedupe (base,ix,iy) so mask count matches jnp scatter ----
  if (tid < NB32) {
    const int mykey = s_key[tid];
    int uni = 1;
    for (int k = 0; k < tid; ++k)
      if (s_key[k] == mykey) { uni = 0; break; }
    s_uni[tid] = uni;
    if (uni) corr = obj * obj;
  }
  __syncthreads();

  int nun = 0;
  for (int k = 0; k < NB32; ++k) nun += s_uni[k];
  const float cnt = (float)(3 * CELLS - nun);

  // ---- Phase 3: label MSE partial: sum over (32 boxes x 80 classes) ----
  float labelp = 0.0f;
  for (int t = tid; t < NB32 * C80; t += 256) {         // exactly 10 iters/thread
    const int j  = t / C80;
    const int cc = t - j * C80;
    const float lab = x[s_key[j] + (5 + cc) * CELLS];
    const float hot = (cc == s_cls[j]) ? 1.0f : 0.0f;
    const float d = lab - hot;
    labelp += d * d;
  }

  // ---- Phase 4: noobj sum of squares over obj channels {0,85,170} ----
  float x2p = 0.0f;
  for (int f = tid; f < 3 * CELLS; f += 256) {
    const int ch = f / CELLS;
    const int p  = f - ch * CELLS;
    const float v = x[ch * 85 * CELLS + p];             // coalesced per segment
    x2p += v * v;
  }
  x2p -= corr;   // remove unique box positions (mask set False there)

  // ---- Phase 5: combine + exact block reduction via WMMA ----
  const float p = boxterm * (5.0f / 32.0f)              // 5 * mean over 32 boxes
                + labelp  * (1.0f / 2560.0f)            // mean over 32*80
                + 0.5f * x2p / cnt;                     // LAMBDA_NOOBJ * noobj

  const float ws = wave_sum_wmma(p);                    // uniform: all 256 active
  const int lane = tid & 31, wave = tid >> 5;
  if (lane == 0) s_ws[wave] = ws;
  __syncthreads();
  if (tid == 0) {
    float L = 0.0f;
    for (int i = 0; i < 8; ++i) L += s_ws[i];
    per_img[img] = L;
  }
}

// Deterministic 256 -> 1 final sum (no float atomics).
__global__ __launch_bounds__(256) void yolo_finalize(
    const float* __restrict__ per_img, float* __restrict__ out)
{
  __shared__ float s_ws[8];
  const float v = per_img[threadIdx.x];
  const float ws = wave_sum_wmma(v);
  const int lane = threadIdx.x & 31, wave = threadIdx.x >> 5;
  if (lane == 0) s_ws[wave] = ws;
  __syncthreads();
  if (threadIdx.x == 0) {
    float L = 0.0f;
    for (int i = 0; i < 8; ++i) L += s_ws[i];
    out[0] = L;
  }
}

extern "C" void kernel_launch(void* const* d_in, const int* in_sizes, int n_in,
                              void* d_out, int out_size, void* d_ws, size_t ws_size,
                              hipStream_t stream) {
  const float* batch_x     = (const float*)d_in[0];  // (256,255,26,26) f32
  const float* batch_box   = (const float*)d_in[1];  // (256,32,5)      f32
  const int*   batch_index = (const int*)  d_in[2];  // (256,32)        i32
  float* out     = (float*)d_out;
  float* per_img = (float*)d_ws;                     // 256 floats of scratch

  yolo_image_loss<<<NIMG, 256, 0, stream>>>(batch_x, batch_box, batch_index, per_img);
  yolo_finalize<<<1, 256, 0, stream>>>(per_img, out);
}